// codeModel3_noC_51960514347246
// MI455X (gfx1250) — compile-verified
//
#include <hip/hip_runtime.h>
#include <hip/hip_bf16.h>
#include <math.h>

// Problem constants (from reference)
#define V_  50257
#define H_  256
#define B_  16
#define L_  2048
#define E_  16384
#define EN_ (E_ + L_)     // edges + self loops per graph
#define G_  (2 * B_)      // 32 graphs total (diff + msg)
#define NEG_SLOPE_ 0.2f

typedef float v2f __attribute__((ext_vector_type(2)));
typedef float v8f __attribute__((ext_vector_type(8)));

// ---- monotonic float <-> uint encoding for atomicMax-based segment_max ----
__device__ __forceinline__ unsigned fenc(float f) {
  unsigned b = __float_as_uint(f);
  return (b & 0x80000000u) ? ~b : (b | 0x80000000u);
}
__device__ __forceinline__ float fdec(unsigned u) {
  unsigned b = (u & 0x80000000u) ? (u & 0x7FFFFFFFu) : ~u;
  return __uint_as_float(b);
}

// ---------------- K0: zero the atomic accumulation buffers ----------------
__global__ void k_zero(float4* __restrict__ p, int n4) {
  int i = blockIdx.x * blockDim.x + threadIdx.x;
  int stride = gridDim.x * blockDim.x;
  float4 z = {0.f, 0.f, 0.f, 0.f};
  for (; i < n4; i += stride) p[i] = z;
}

// ---------------- K1: xl = gather(emb) @ Wl + bl ; xr = ... @ Wr + br -----
// One wave per 16x16 output tile; computes BOTH xl and xr tiles (A reuse).
// f32 WMMA 16x16x4: full reference precision.
__global__ void k_gat_linear(const int* __restrict__ diff_ids,
                             const int* __restrict__ msg_ids,
                             const float* __restrict__ emb,
                             const float* __restrict__ Wl, const float* __restrict__ bl,
                             const float* __restrict__ Wr, const float* __restrict__ br,
                             float* __restrict__ xl, float* __restrict__ xr) {
  const int g    = blockIdx.y;                       // graph 0..31
  const int lane = threadIdx.x & 31;
  const int wave = threadIdx.x >> 5;
  const int tile = blockIdx.x * 4 + wave;            // 0..2047  (128 Mtiles x 16 Ntiles)
  const int m0   = (tile >> 4) * 16;
  const int n0   = (tile & 15) * 16;

  const int* ids = (g < B_) ? (diff_ids + g * L_) : (msg_ids + (g - B_) * L_);

  // A fragment (16x4 f32): lanes 0-15 hold K=k0..k0+1, lanes 16-31 hold K=k0+2..k0+3
  const int   row  = m0 + (lane & 15);
  const float* arow = emb + (long)ids[row] * H_;     // gathered embedding row
  const int   koff = (lane >> 4) * 2;                // 0 or 2
  const int   col  = n0 + (lane & 15);               // output column

  v8f cl = {};
  v8f cr = {};
  for (int k0 = 0; k0 < H_; k0 += 4) {
    v2f a = *(const v2f*)(arow + k0 + koff);         // 8B-aligned (k0+koff even)
    v2f bfl, bfr;
    bfl[0] = Wl[(k0 + koff)     * H_ + col];
    bfl[1] = Wl[(k0 + koff + 1) * H_ + col];
    bfr[0] = Wr[(k0 + koff)     * H_ + col];
    bfr[1] = Wr[(k0 + koff + 1) * H_ + col];
    cl = __builtin_amdgcn_wmma_f32_16x16x4_f32(false, a, false, bfl, (short)0, cl, false, false);
    cr = __builtin_amdgcn_wmma_f32_16x16x4_f32(false, a, false, bfr, (short)0, cr, false, false);
  }

  // C/D layout: VGPR r -> row m0+r (lanes 0-15) / m0+8+r (lanes 16-31), col = n0+(lane&15)
  const float blv = bl[col], brv = br[col];
  const int rbase = m0 + ((lane >> 4) << 3);
  float* xlo = xl + (long)g * L_ * H_;
  float* xro = xr + (long)g * L_ * H_;
#pragma unroll
  for (int r = 0; r < 8; ++r) {
    xlo[(long)(rbase + r) * H_ + col] = cl[r] + blv;
    xro[(long)(rbase + r) * H_ + col] = cr[r] + brv;
  }
}

// ---------------- K2: per-edge attention score + segment max --------------
// One wave per edge: e = leakyrelu(xl[src]+xr[dst]) . att  (256-dot)
__global__ void k_edge_score(const int* __restrict__ ei_diff,
                             const int* __restrict__ ei_msg,
                             const float* __restrict__ xl, const float* __restrict__ xr,
                             const float* __restrict__ att,
                             float* __restrict__ e_buf, unsigned* __restrict__ m_enc) {
  const int g    = blockIdx.y;
  const int lane = threadIdx.x & 31;
  const int idx  = blockIdx.x * 8 + (threadIdx.x >> 5);  // 0..EN_-1
  const int* ei  = (g < B_) ? (ei_diff + (long)g * 2 * E_)
                            : (ei_msg + (long)(g - B_) * 2 * E_);
  int src, dst;
  if (idx < E_) { src = ei[idx]; dst = ei[E_ + idx]; }
  else          { src = dst = idx - E_; }                // self loop

  const float* pl = xl + ((long)g * L_ + src) * H_;
  const float* pr = xr + ((long)g * L_ + dst) * H_;
  float s = 0.f;
  for (int j = lane; j < H_; j += 32) {
    float v = pl[j] + pr[j];
    v = (v >= 0.f) ? v : NEG_SLOPE_ * v;
    s += v * att[j];
  }
  for (int off = 16; off > 0; off >>= 1) s += __shfl_down(s, off, 32);
  if (lane == 0) {
    e_buf[(long)g * EN_ + idx] = s;
    atomicMax(m_enc + g * L_ + dst, fenc(s));
  }
}

// ---------------- K3: a = exp(e - m[dst]); denom[dst] += a ----------------
__global__ void k_edge_exp(const int* __restrict__ ei_diff,
                           const int* __restrict__ ei_msg,
                           float* __restrict__ e_buf,
                           const unsigned* __restrict__ m_enc,
                           float* __restrict__ denom) {
  const int g   = blockIdx.y;
  const int idx = blockIdx.x * 256 + threadIdx.x;
  const int* ei = (g < B_) ? (ei_diff + (long)g * 2 * E_)
                           : (ei_msg + (long)(g - B_) * 2 * E_);
  const int dst = (idx < E_) ? ei[E_ + idx] : (idx - E_);
  const float m = fdec(m_enc[g * L_ + dst]);
  const float a = expf(e_buf[(long)g * EN_ + idx] - m);
  e_buf[(long)g * EN_ + idx] = a;
  atomicAdd(denom + g * L_ + dst, a);
}

// ---------------- K4: out[dst] += (a/denom[dst]) * xl[src] ----------------
__global__ void k_aggregate(const int* __restrict__ ei_diff,
                            const int* __restrict__ ei_msg,
                            const float* __restrict__ xl,
                            const float* __restrict__ e_buf,
                            const float* __restrict__ denom,
                            float* __restrict__ out_agg) {
  const int g    = blockIdx.y;
  const int lane = threadIdx.x & 31;
  const int idx  = blockIdx.x * 8 + (threadIdx.x >> 5);
  const int* ei  = (g < B_) ? (ei_diff + (long)g * 2 * E_)
                            : (ei_msg + (long)(g - B_) * 2 * E_);
  int src, dst;
  if (idx < E_) { src = ei[idx]; dst = ei[E_ + idx]; }
  else          { src = dst = idx - E_; }

  float alpha = 0.f;
  if (lane == 0)
    alpha = e_buf[(long)g * EN_ + idx] / denom[g * L_ + dst];
  alpha = __shfl(alpha, 0, 32);

  const float* pl = xl + ((long)g * L_ + src) * H_;
  float* po = out_agg + ((long)g * L_ + dst) * H_;
  for (int j = lane; j < H_; j += 32) atomicAdd(po + j, alpha * pl[j]);
}

// ---------------- K5: fused[b,h] = max over both graphs & nodes -----------
// residual: (gat_out + gat_bias) + x, then max over 2L node axis
__global__ void k_maxreduce(const int* __restrict__ diff_ids,
                            const int* __restrict__ msg_ids,
                            const float* __restrict__ emb,
                            const float* __restrict__ out_agg,
                            const float* __restrict__ gat_bias,
                            float* __restrict__ fused) {
  const int b = blockIdx.x;     // 0..15
  const int h = threadIdx.x;    // 0..255
  const float gb = gat_bias[h];
  float acc = -INFINITY;
  for (int half = 0; half < 2; ++half) {
    const int g = b + half * B_;
    const int* ids = half ? (msg_ids + b * L_) : (diff_ids + b * L_);
    const float* og = out_agg + (long)g * L_ * H_;
    for (int n = 0; n < L_; ++n) {
      const float xv = emb[(long)ids[n] * H_ + h];
      acc = fmaxf(acc, og[(long)n * H_ + h] + gb + xv);
    }
  }
  fused[b * H_ + h] = acc;
}

// ---------------- K6a/b: MLP layers with f32 WMMA (M=16 fits exactly) -----
// h_out = relu(A(16xK) @ W(KxNout) + b), one wave per 16-wide N tile.
template <int KDIM, int NOUT>
__global__ void k_mlp_wmma(const float* __restrict__ A, const float* __restrict__ W,
                           const float* __restrict__ bias, float* __restrict__ out) {
  const int lane = threadIdx.x & 31;
  const int wave = threadIdx.x >> 5;
  const int n0   = wave * 16;
  const float* arow = A + (lane & 15) * KDIM;
  const int koff = (lane >> 4) * 2;
  const int col  = n0 + (lane & 15);
  v8f c = {};
  for (int k0 = 0; k0 < KDIM; k0 += 4) {
    v2f a = *(const v2f*)(arow + k0 + koff);
    v2f bf;
    bf[0] = W[(k0 + koff)     * NOUT + col];
    bf[1] = W[(k0 + koff + 1) * NOUT + col];
    c = __builtin_amdgcn_wmma_f32_16x16x4_f32(false, a, false, bf, (short)0, c, false, false);
  }
  const float bv = bias[col];
  const int rbase = (lane >> 4) << 3;
#pragma unroll
  for (int r = 0; r < 8; ++r)
    out[(rbase + r) * NOUT + col] = fmaxf(c[r] + bv, 0.f);
}

// ---------------- K6c: final 128 -> 2 layer (tiny, scalar) ----------------
__global__ void k_mlp2(const float* __restrict__ h1, const float* __restrict__ W2,
                       const float* __restrict__ b2, float* __restrict__ out) {
  const int t = threadIdx.x;
  if (t >= 32) return;
  const int b = t >> 1, c = t & 1;
  float s = b2[c];
  for (int k = 0; k < 128; ++k) s += h1[b * 128 + k] * W2[k * 2 + c];
  out[b * 2 + c] = s;
}

extern "C" void kernel_launch(void* const* d_in, const int* in_sizes, int n_in,
                              void* d_out, int out_size, void* d_ws, size_t ws_size,
                              hipStream_t stream) {
  (void)in_sizes; (void)n_in; (void)out_size; (void)ws_size;
  const int*   diff_ids = (const int*)  d_in[0];
  const int*   msg_ids  = (const int*)  d_in[1];
  const int*   ei_diff  = (const int*)  d_in[2];
  const int*   ei_msg   = (const int*)  d_in[3];
  const float* emb      = (const float*)d_in[4];
  const float* Wl       = (const float*)d_in[5];
  const float* bl       = (const float*)d_in[6];
  const float* Wr       = (const float*)d_in[7];
  const float* br       = (const float*)d_in[8];
  const float* att      = (const float*)d_in[9];
  const float* gat_bias = (const float*)d_in[10];
  const float* W0       = (const float*)d_in[11];
  const float* b0       = (const float*)d_in[12];
  const float* W1       = (const float*)d_in[13];
  const float* b1       = (const float*)d_in[14];
  const float* W2       = (const float*)d_in[15];
  const float* b2       = (const float*)d_in[16];
  float* outp = (float*)d_out;

  // Workspace layout (~196 MiB). out_agg / m_enc / denom kept contiguous
  // so one zero kernel clears all atomic accumulators every call.
  char* w = (char*)d_ws;
  const size_t xsz = (size_t)G_ * L_ * H_ * sizeof(float);   // 64 MiB each
  float*    xl      = (float*)w;                 w += xsz;
  float*    xr      = (float*)w;                 w += xsz;
  float*    out_agg = (float*)w;                 w += xsz;
  unsigned* m_enc   = (unsigned*)w;              w += (size_t)G_ * L_ * sizeof(unsigned);
  float*    denom   = (float*)w;                 w += (size_t)G_ * L_ * sizeof(float);
  float*    e_buf   = (float*)w;                 w += (size_t)G_ * EN_ * sizeof(float);
  float*    fused   = (float*)w;                 w += (size_t)B_ * H_ * sizeof(float);
  float*    h0      = (float*)w;                 w += (size_t)B_ * 512 * sizeof(float);
  float*    h1      = (float*)w;                 w += (size_t)B_ * 128 * sizeof(float);

  // K0: zero out_agg + m_enc + denom (contiguous, counts all multiples of 4)
  const int n4 = (G_ * L_ * H_ + 2 * G_ * L_) / 4;
  k_zero<<<4096, 256, 0, stream>>>((float4*)out_agg, n4);

  // K1: xl/xr GEMMs (WMMA f32). 2048 tiles/graph, 4 tiles/block.
  k_gat_linear<<<dim3(512, G_), 128, 0, stream>>>(diff_ids, msg_ids, emb,
                                                  Wl, bl, Wr, br, xl, xr);

  // K2..K4: edge softmax + aggregation
  k_edge_score<<<dim3(EN_ / 8, G_), 256, 0, stream>>>(ei_diff, ei_msg, xl, xr, att,
                                                      e_buf, m_enc);
  k_edge_exp<<<dim3(EN_ / 256, G_), 256, 0, stream>>>(ei_diff, ei_msg, e_buf, m_enc, denom);
  k_aggregate<<<dim3(EN_ / 8, G_), 256, 0, stream>>>(ei_diff, ei_msg, xl, e_buf, denom,
                                                     out_agg);

  // K5: residual + global max pool over 2L nodes
  k_maxreduce<<<B_, H_, 0, stream>>>(diff_ids, msg_ids, emb, out_agg, gat_bias, fused);

  // K6: MLP head (WMMA for the two big layers)
  k_mlp_wmma<256, 512><<<1, 1024, 0, stream>>>(fused, W0, b0, h0);  // 32 waves
  k_mlp_wmma<512, 128><<<1, 256,  0, stream>>>(h0,    W1, b1, h1);  // 8 waves
  k_mlp2<<<1, 32, 0, stream>>>(h1, W2, b2, outp);
}